// DetectionOutput_2345052143762
// MI455X (gfx1250) — compile-verified
//
#include <hip/hip_runtime.h>
#include <stdint.h>
#include <stddef.h>

// ---------------- problem constants (match reference) ----------------
#define B_          4
#define N_          2000
#define NPAD        2048          // pow2 pad for bitonic sort
#define C_          81
#define CM          80            // foreground classes
#define MAXDET      100
#define CAND_PER_B  (CM * MAXDET) // 8000 candidates per batch
#define TOPM        8192          // pow2 pad for per-batch top-k sort
#define NMS_THR_    0.7f
#define SCORE_THR_  0.01f
#define MAX_LOG_WH_ 4.135166556742356f  // log(1000/16)

#define CHUNK       400           // TDM tile rows per transfer (5 * 400 = N_)
#define NCHUNK      (N_ / CHUNK)

typedef __attribute__((ext_vector_type(4))) unsigned int u32x4;
typedef __attribute__((ext_vector_type(8))) unsigned int u32x8;

__device__ __forceinline__ float neg_inf() { return __int_as_float(0xff800000); }

// Generic pointers to LDS carry the wave-relative LDS byte offset in their
// low 32 bits on AMDGPU — exactly what the async/TDM LDS address fields want.
__device__ __forceinline__ uint32_t lds_off(const void* p) {
  return (uint32_t)(uintptr_t)p;
}

// ---- CDNA5 async global->LDS (ASYNCcnt-tracked) -----------------------------
__device__ __forceinline__ void async_ld_b128(uint32_t lds_dst, const void* gsrc) {
  asm volatile("global_load_async_to_lds_b128 %0, %1, off"
               :: "v"(lds_dst), "v"((unsigned long long)(uintptr_t)gsrc)
               : "memory");
}
__device__ __forceinline__ void async_ld_b32(uint32_t lds_dst, const void* gsrc) {
  asm volatile("global_load_async_to_lds_b32 %0, %1, off"
               :: "v"(lds_dst), "v"((unsigned long long)(uintptr_t)gsrc)
               : "memory");
}
__device__ __forceinline__ void wait_async_0() {
  asm volatile("s_wait_asynccnt 0x0" ::: "memory");
}

// ---- CDNA5 Tensor Data Mover: strided 2D tile -> contiguous LDS -------------
// D# per CDNA5 ISA (2D tensor: groups 0+1 only).
__device__ __forceinline__ void tdm_load_2d(uint32_t lds_dst_bytes,
                                            const void* gtile,      // tile start
                                            uint32_t tensor_d0,     // elems (>= tile_d0)
                                            uint32_t tensor_d1,     // rows  (>= rows)
                                            uint32_t tile_d0,       // elems per row
                                            uint32_t rows,          // tile_dim1
                                            uint32_t row_stride) {  // elems
  const unsigned long long ga = (unsigned long long)(uintptr_t)gtile;
  u32x4 g0;
  g0[0] = 1u;                                            // count=1 (valid), user mode
  g0[1] = lds_dst_bytes;                                 // lds_addr
  g0[2] = (uint32_t)(ga & 0xFFFFFFFFull);                // global_addr[31:0]
  g0[3] = (uint32_t)((ga >> 32) & 0x01FFFFFFull)         // global_addr[56:32]
          | (2u << 30);                                  // type=2 ("image")
  u32x8 g1;
  g1[0] = (2u << 16);                                    // data_size=2 -> 4B elems
  g1[1] = (tensor_d0 & 0xFFFFu) << 16;                   // tensor_dim0[15:0]
  g1[2] = (tensor_d0 >> 16) | ((tensor_d1 & 0xFFFFu) << 16);
  g1[3] = (tensor_d1 >> 16) | (tile_d0 << 16);           // tile_dim0
  g1[4] = rows;                                          // tile_dim1 (tile_dim2=0)
  g1[5] = row_stride;                                    // tensor_dim0_stride[31:0]
  g1[6] = 0u;                                            // stride hi / dim1_stride lo
  g1[7] = 0u;
  asm volatile("tensor_load_to_lds %0, %1" :: "s"(g0), "s"(g1) : "memory");
}

// ---- in-LDS bitonic sort, descending by key, ties -> ascending index --------
template <int M, int NT, typename IT>
__device__ __forceinline__ void bitonic_sort_desc(float* key, IT* idx, int tid) {
  for (int k = 2; k <= M; k <<= 1) {
    for (int j = k >> 1; j > 0; j >>= 1) {
      __syncthreads();
      for (int i = tid; i < M; i += NT) {
        const int l = i ^ j;
        if (l > i) {
          const float ki = key[i], kl = key[l];
          const IT    ii = idx[i], il = idx[l];
          const bool up = ((i & k) == 0);
          const bool l_first = (kl > ki) || (kl == ki && il < ii);
          const bool i_first = (ki > kl) || (ki == kl && ii < il);
          if (up ? l_first : i_first) {
            key[i] = kl; key[l] = ki;
            idx[i] = il; idx[l] = ii;
          }
        }
      }
    }
  }
  __syncthreads();
}

// =============================================================================
// Kernel A: one workgroup per (batch, fg-class).
//   async-stage rois+scores into LDS; double-buffered TDM pipeline stages the
//   strided delta tiles (DMA overlapped with decode via tensorcnt<=1);
//   decode+clip, bitonic sort by score, greedy NMS scan, compact top-100.
// =============================================================================
__global__ __launch_bounds__(256) void det_per_class(
    const float* __restrict__ rois,      // [B,N,4]
    const float* __restrict__ deltas,    // [B*N, C*4]
    const float* __restrict__ scores,    // [B*N, C]
    const float* __restrict__ im_info,   // [B,3]
    float*  __restrict__ cand_sc,        // [B, CM*MAXDET]
    float4* __restrict__ cand_box) {     // [B, CM*MAXDET]
  constexpr int NT = 256;
  const int b   = blockIdx.x / CM;
  const int cm  = blockIdx.x % CM;
  const int cls = cm + 1;                // original class index (skip bg 0)
  const int tid = threadIdx.x;

  __shared__ float4        s_box[NPAD];        // rois -> decoded boxes (32 KB)
  __shared__ float         s_key[NPAD];        // scores -> sort keys   ( 8 KB)
  __shared__ int           s_idx[NPAD];        //                       ( 8 KB)
  __shared__ unsigned char s_keep[NPAD];       //                       ( 2 KB)
  __shared__ float4        s_delta[2][CHUNK];  // TDM double buffer     (12.5 KB)

  // ---- async stage: rois row (B128) + this class's score column (B32) ----
  const float* rois_b = rois + (size_t)b * N_ * 4;
  for (int it = 0; it < (NPAD + NT - 1) / NT; ++it) {
    const int n = tid + it * NT;
    if (n < N_) {
      async_ld_b128(lds_off(&s_box[n]), rois_b + (size_t)n * 4);
      async_ld_b32 (lds_off(&s_key[n]), scores + (size_t)(b * N_ + n) * C_ + cls);
    }
  }

  // ---- prime TDM pipeline (wave 0 only; TDM ignores EXEC) ----
  if (tid < 32) {
    tdm_load_2d(lds_off(&s_delta[0][0]),
                deltas + (size_t)(b * N_) * (C_ * 4) + cls * 4,
                /*tensor_d0=*/C_ * 4, /*tensor_d1=*/N_,
                /*tile_d0=*/4, /*rows=*/CHUNK, /*row_stride=*/C_ * 4);
  }

  wait_async_0();
  __syncthreads();

  const float Wm = im_info[b * 3 + 1] - 1.0f;
  const float Hm = im_info[b * 3 + 0] - 1.0f;

  // init indices + pad keys beyond N_
  for (int n = tid; n < NPAD; n += NT) {
    s_idx[n] = n;
    if (n >= N_) s_key[n] = neg_inf();
  }

  // ---- double-buffered TDM decode pipeline ----
  for (int c = 0; c < NCHUNK; ++c) {
    if (tid < 32) {
      if (c + 1 < NCHUNK) {
        // issue next tile into the buffer drained at the end of iter c-1,
        // then wait for the *older* of the two in-flight tiles (in-order).
        tdm_load_2d(lds_off(&s_delta[(c + 1) & 1][0]),
                    deltas + (size_t)(b * N_ + (c + 1) * CHUNK) * (C_ * 4) + cls * 4,
                    C_ * 4, N_, 4, CHUNK, C_ * 4);
        __builtin_amdgcn_s_wait_tensorcnt(1);
      } else {
        __builtin_amdgcn_s_wait_tensorcnt(0);   // drain last tile
      }
    }
    __syncthreads();                            // tile c visible to all waves

    for (int r = tid; r < CHUNK; r += NT) {
      const int n = c * CHUNK + r;
      const float4 rr = s_box[n];
      const float w  = rr.z - rr.x + 1.0f;
      const float h  = rr.w - rr.y + 1.0f;
      const float cx = rr.x + 0.5f * w;
      const float cy = rr.y + 0.5f * h;
      const float4 d = s_delta[c & 1][r];
      const float dx = d.x * 0.1f;
      const float dy = d.y * 0.1f;
      const float dw = fminf(d.z * 0.2f, MAX_LOG_WH_);
      const float dh = fminf(d.w * 0.2f, MAX_LOG_WH_);
      const float pcx = dx * w + cx;
      const float pcy = dy * h + cy;
      const float pw  = expf(dw) * w;
      const float ph  = expf(dh) * h;
      const float bx1 = fminf(fmaxf(pcx - 0.5f * pw, 0.0f), Wm);
      const float by1 = fminf(fmaxf(pcy - 0.5f * ph, 0.0f), Hm);
      const float bx2 = fminf(fmaxf(pcx + 0.5f * pw - 1.0f, 0.0f), Wm);
      const float by2 = fminf(fmaxf(pcy + 0.5f * ph - 1.0f, 0.0f), Hm);
      s_box[n] = make_float4(bx1, by1, bx2, by2);
      const float sc = s_key[n];
      s_key[n] = (sc > SCORE_THR_) ? sc : neg_inf();
    }
    __syncthreads();   // buffer (c&1) fully consumed before reuse at c+2
  }

  // ---- sort (score desc, stable via index tiebreak) ----
  bitonic_sort_desc<NPAD, NT, int>(s_key, s_idx, tid);

  for (int n = tid; n < NPAD; n += NT)
    s_keep[n] = (s_key[n] > -3.0e38f) ? 1 : 0;   // valid_s after sort

  // ---- greedy NMS scan (matches reference lax.scan semantics) ----
  for (int i = 0; i < N_ - 1; ++i) {
    __syncthreads();
    if (!s_keep[i]) continue;                    // uniform branch across block
    const float4 bi = s_box[s_idx[i]];
    const float  ai = (bi.z - bi.x + 1.0f) * (bi.w - bi.y + 1.0f);
    for (int j = i + 1 + tid; j < N_; j += NT) {
      if (!s_keep[j]) continue;
      const float4 bj = s_box[s_idx[j]];
      const float xx1 = fmaxf(bi.x, bj.x);
      const float yy1 = fmaxf(bi.y, bj.y);
      const float xx2 = fminf(bi.z, bj.z);
      const float yy2 = fminf(bi.w, bj.w);
      const float iw  = fmaxf(xx2 - xx1 + 1.0f, 0.0f);
      const float ih  = fmaxf(yy2 - yy1 + 1.0f, 0.0f);
      const float inter = iw * ih;
      const float aj = (bj.z - bj.x + 1.0f) * (bj.w - bj.y + 1.0f);
      const float iou = inter / (ai + aj - inter);
      if (iou > NMS_THR_) s_keep[j] = 0;
    }
  }
  __syncthreads();

  // ---- compact first <=100 kept (in score order) to global candidates ----
  if (tid == 0) {
    const size_t base = ((size_t)b * CM + cm) * MAXDET;
    int cnt = 0;
    for (int j = 0; j < N_ && cnt < MAXDET; ++j) {
      if (s_keep[j]) {
        cand_sc[base + cnt]  = s_key[j];
        cand_box[base + cnt] = s_box[s_idx[j]];
        ++cnt;
      }
    }
    for (; cnt < MAXDET; ++cnt) {
      cand_sc[base + cnt]  = neg_inf();
      cand_box[base + cnt] = make_float4(0.0f, 0.0f, 0.0f, 0.0f);
    }
  }
}

// =============================================================================
// Kernel B: one workgroup per batch — top-100 of the 8000 per-class survivors.
// =============================================================================
__global__ __launch_bounds__(1024) void topk_batch(
    const float*  __restrict__ cand_sc,
    const float4* __restrict__ cand_box,
    float* __restrict__ out) {
  constexpr int NT = 1024;
  const int b   = blockIdx.x;
  const int tid = threadIdx.x;

  __shared__ float          k_key[TOPM];  // 32 KB
  __shared__ unsigned short k_idx[TOPM];  // 16 KB

  for (int s = tid; s < TOPM; s += NT) {
    k_key[s] = (s < CAND_PER_B) ? cand_sc[(size_t)b * CAND_PER_B + s] : neg_inf();
    k_idx[s] = (unsigned short)s;
  }
  bitonic_sort_desc<TOPM, NT, unsigned short>(k_key, k_idx, tid);

  if (tid < MAXDET) {
    const float sc  = k_key[tid];
    const int  slot = (int)k_idx[tid];
    const bool ok   = sc > -3.0e38f;     // isfinite(top_sc)
    const float4 bx = ok ? cand_box[(size_t)b * CAND_PER_B + slot]
                         : make_float4(0.0f, 0.0f, 0.0f, 0.0f);
    const int cls_o = ok ? (slot / MAXDET + 1) : 0;
    const int o = b * MAXDET + tid;

    float* boxes_out  = out;                                   // [B*100, 4]
    int*   class_out  = (int*)(out + (size_t)B_ * MAXDET * 4); // [B*100]
    float* score_out  = out + (size_t)B_ * MAXDET * 5;         // [B*100]
    int*   batch_out  = (int*)(out + (size_t)B_ * MAXDET * 6); // [B*100]

    boxes_out[o * 4 + 0] = bx.x;
    boxes_out[o * 4 + 1] = bx.y;
    boxes_out[o * 4 + 2] = bx.z;
    boxes_out[o * 4 + 3] = bx.w;
    class_out[o] = cls_o;
    score_out[o] = ok ? sc : 0.0f;
    batch_out[o] = b;
  }
}

extern "C" void kernel_launch(void* const* d_in, const int* in_sizes, int n_in,
                              void* d_out, int out_size, void* d_ws, size_t ws_size,
                              hipStream_t stream) {
  (void)in_sizes; (void)n_in; (void)out_size; (void)ws_size;

  const float* all_rois       = (const float*)d_in[0];  // [B,N,4]
  const float* all_box_deltas = (const float*)d_in[1];  // [B*N, C*4]
  const float* all_cls_scores = (const float*)d_in[2];  // [B*N, C]
  const float* im_info        = (const float*)d_in[3];  // [B,3]

  // Workspace layout (<= 640 KB): per-class survivor scores then boxes.
  float*  cand_sc  = (float*)d_ws;                                   // B*8000 f32
  float4* cand_box = (float4*)((char*)d_ws +
                     (size_t)B_ * CAND_PER_B * sizeof(float));       // B*8000 f32x4

  det_per_class<<<dim3(B_ * CM), dim3(256), 0, stream>>>(
      all_rois, all_box_deltas, all_cls_scores, im_info, cand_sc, cand_box);

  topk_batch<<<dim3(B_), dim3(1024), 0, stream>>>(
      cand_sc, cand_box, (float*)d_out);
}